// RadialNetwork2d_75299366633508
// MI455X (gfx1250) — compile-verified
//
#include <hip/hip_runtime.h>

typedef __attribute__((ext_vector_type(2))) float v2f;
typedef __attribute__((ext_vector_type(4))) float v4f;
typedef __attribute__((ext_vector_type(8))) float v8f;

// Reference constants: VAR = 0.25, NORM = 1/(2*pi*VAR) = 2/pi
#define NORM_F     0.63661977236758138f
// basis = NORM * exp(-0.5*sq/VAR) = NORM * exp2(sq * (-2 * log2(e)))
#define EXP2_COEF  -2.88539008177792681f
#define NCENTERS   1600
#define NBLOCKS    100   // 1600 / 16

#if !__has_builtin(__builtin_amdgcn_exp2f)
#define __builtin_amdgcn_exp2f exp2f
#endif

// One wave32 computes out[rowBase .. rowBase+15][0..3].
//
// WMMA trick: with A-row m = [px, py, ||p||^2, 1] (16x4) and
// B-col n = [-2cx, -2cy, 1, ||c||^2] (4x16), the 16x16x4 f32 WMMA yields
// D[m,n] = ||p_m - c_n||^2 in one instruction per 16x16 tile.
__global__ __launch_bounds__(256) void radial_wmma_kernel(
    const float* __restrict__ pos,   // [B,2]
    const float* __restrict__ cen,   // [1600,2]
    const float* __restrict__ Wm,    // [1600,4]
    const float* __restrict__ bias,  // [4]
    float* __restrict__ out,         // [B,4]
    int nrows)
{
    const int lane    = threadIdx.x & 31;
    const int wave    = (blockIdx.x * blockDim.x + threadIdx.x) >> 5;
    const int rowBase = wave * 16;
    if (rowBase >= nrows) return;               // wave-uniform exit

    const int  m      = lane & 15;              // tile row (A) / tile col (B,D)
    const bool hiHalf = (lane >> 4) != 0;

    // ---- Build loop-invariant A operand (32-bit A 16x4 layout):
    // lanes 0-15 : VGPR0 = K0 (px),   VGPR1 = K1 (py)
    // lanes 16-31: VGPR0 = K2 (||p||^2), VGPR1 = K3 (1.0)
    v2f pv = *(const v2f*)(pos + (size_t)(rowBase + m) * 2);
    const float pp = pv.x * pv.x + pv.y * pv.y;
    v2f A;
    A.x = hiHalf ? pp   : pv.x;
    A.y = hiHalf ? 1.0f : pv.y;

    float acc[8][4];
#pragma unroll
    for (int v = 0; v < 8; ++v)
#pragma unroll
        for (int a = 0; a < 4; ++a) acc[v][a] = 0.0f;

    for (int blk = 0; blk < NBLOCKS; ++blk) {
        const int n = blk * 16 + m;

        // ---- B operand (32-bit B 4x16 layout, mirror of A):
        // lanes 0-15 : VGPR0 = K0 (-2cx), VGPR1 = K1 (-2cy)
        // lanes 16-31: VGPR0 = K2 (1.0),  VGPR1 = K3 (||c||^2)
        v2f cv = *(const v2f*)(cen + (size_t)n * 2);
        const float cc = cv.x * cv.x + cv.y * cv.y;
        v2f Bv;
        Bv.x = hiHalf ? 1.0f : -2.0f * cv.x;
        Bv.y = hiHalf ? cc   : -2.0f * cv.y;

        // D = A x B + 0  ->  squared distances for a 16x16 tile
        v8f D = {0.f, 0.f, 0.f, 0.f, 0.f, 0.f, 0.f, 0.f};
        D = __builtin_amdgcn_wmma_f32_16x16x4_f32(
                /*neg_a=*/false, A, /*neg_b=*/false, Bv,
                /*c_mod=*/(short)0, D, /*reuse_a=*/false, /*reuse_b=*/false);

        // This lane owns column n of the tile -> contract against W[n][0..3].
        const v4f w = *(const v4f*)(Wm + (size_t)n * 4);
#pragma unroll
        for (int v = 0; v < 8; ++v) {
            const float basis = NORM_F * __builtin_amdgcn_exp2f(D[v] * EXP2_COEF);
            acc[v][0] = __builtin_fmaf(basis, w.x, acc[v][0]);
            acc[v][1] = __builtin_fmaf(basis, w.y, acc[v][1]);
            acc[v][2] = __builtin_fmaf(basis, w.z, acc[v][2]);
            acc[v][3] = __builtin_fmaf(basis, w.w, acc[v][3]);
        }
    }

    // ---- Reduce over the 16 columns held by each half-wave.
    // D layout: VGPR v / lanes 0-15 -> row v ; lanes 16-31 -> row v+8.
    // xor masks 1,2,4,8 keep the butterfly within each 16-lane half.
    const v4f bv = *(const v4f*)(bias);
#pragma unroll
    for (int v = 0; v < 8; ++v) {
#pragma unroll
        for (int a = 0; a < 4; ++a) {
            float s = acc[v][a];
            s += __shfl_xor(s, 1, 32);
            s += __shfl_xor(s, 2, 32);
            s += __shfl_xor(s, 4, 32);
            s += __shfl_xor(s, 8, 32);
            acc[v][a] = s;
        }
        if (m == v) {   // 2 active lanes write row v (half0) and row v+8 (half1)
            const int row = rowBase + v + (hiHalf ? 8 : 0);
            v4f o;
            o.x = acc[v][0] + bv.x;
            o.y = acc[v][1] + bv.y;
            o.z = acc[v][2] + bv.z;
            o.w = acc[v][3] + bv.w;
            *(v4f*)(out + (size_t)row * 4) = o;
        }
    }
}

extern "C" void kernel_launch(void* const* d_in, const int* in_sizes, int n_in,
                              void* d_out, int out_size, void* d_ws, size_t ws_size,
                              hipStream_t stream) {
    const float* pos = (const float*)d_in[0];   // [B,2]
    const float* cen = (const float*)d_in[1];   // [1600,2]
    const float* Wm  = (const float*)d_in[2];   // [1600,4]
    const float* b   = (const float*)d_in[3];   // [4]
    float* out = (float*)d_out;                 // [B,4]

    const int nrows = in_sizes[0] / 2;          // 65536
    const int waves = (nrows + 15) / 16;        // 16 rows per wave
    const int blocks = (waves + 7) / 8;         // 8 wave32s per 256-thread block

    radial_wmma_kernel<<<blocks, 256, 0, stream>>>(pos, cen, Wm, b, out, nrows);
}